// CrystalDecoder_79637283602854
// MI455X (gfx1250) — compile-verified
//
#include <hip/hip_runtime.h>

#define NN 65536
#define EE 524288
#define BB 256
#define LL 64
#define HH 128
#define DD2 256
#define NEF 1024

typedef __attribute__((ext_vector_type(2))) float v2f;
typedef __attribute__((ext_vector_type(8))) float v8f;
typedef __attribute__((ext_vector_type(4))) int v4i;
typedef __attribute__((ext_vector_type(8))) int v8i;
typedef __attribute__((ext_vector_type(4))) unsigned int v4u;

typedef __attribute__((address_space(1))) v4i gv4i;   // global (AS1)
typedef __attribute__((address_space(3))) v4i lv4i;   // LDS (AS3)

// ---------------------------------------------------------------------------
// Async global->LDS staging (CDNA5 GLOBAL_LOAD_ASYNC_TO_LDS_B128, ASYNCcnt).
// ---------------------------------------------------------------------------
#if __has_builtin(__builtin_amdgcn_global_load_async_to_lds_b128)
#define ASYNC_LDS_MODE 1
#elif __has_builtin(__builtin_amdgcn_load_to_lds)
#define ASYNC_LDS_MODE 2
#else
#define ASYNC_LDS_MODE 0
#endif

#if ASYNC_LDS_MODE != 0
#if __has_builtin(__builtin_amdgcn_s_wait_asynccnt)
#define WAIT_ASYNC(n) __builtin_amdgcn_s_wait_asynccnt(n)
#else
#define WAIT_ASYNC(n) asm volatile("s_wait_asynccnt %0" ::"n"(n) : "memory")
#endif
#else
#define WAIT_ASYNC(n) do {} while (0)
#endif

// Tensor Data Mover (TDM) for strided 2D A-tiles (TENSORcnt).
#if __has_builtin(__builtin_amdgcn_tensor_load_to_lds) && \
    __has_builtin(__builtin_amdgcn_s_wait_tensorcnt)
#define HAVE_TDM 1
#else
#define HAVE_TDM 0
#endif

__device__ __forceinline__ void cp16_async(const float* __restrict__ g,
                                           float* l) {
#if ASYNC_LDS_MODE == 1
  __builtin_amdgcn_global_load_async_to_lds_b128((gv4i*)g, (lv4i*)l, 0, 0);
#elif ASYNC_LDS_MODE == 2
  __builtin_amdgcn_load_to_lds((void*)g, (lv4i*)l, 16, 0, 0);
#else
  *reinterpret_cast<float4*>(l) = *reinterpret_cast<const float4*>(g);
#endif
}

__device__ __forceinline__ unsigned lds_off(void* p) {
  return (unsigned)(size_t)(__attribute__((address_space(3))) void*)p;
}

#if HAVE_TDM
// Issue one TDM descriptor: load 64 x 16 fp32 tile from a row-major tensor
// (row stride 1024 elems) into LDS at lds_byte, padding the LDS destination
// by 4 DWORDs after every 16 DWORDs (=> LDS row stride 20 floats,
// bank-conflict free).
__device__ __forceinline__ void tdm_load_tile_a(const float* gsrc,
                                                unsigned lds_byte) {
  unsigned long long ga = (unsigned long long)(size_t)gsrc;
  v4u g0;
  g0.x = 1u;                     // count=1 valid user descriptor
  g0.y = lds_byte;               // lds_addr
  g0.z = (unsigned)ga;           // global_addr[31:0]
  g0.w = (unsigned)((ga >> 32) & 0x01FFFFFFull) | (2u << 30);   // [56:32]|type=2
  v8i g1;
  // data_size=4B (code 2), pad_enable, pad_interval=16DW (code 3),
  // pad_amount=4DW (code 3)
  g1[0] = (int)((2u << 16) | (1u << 20) | (3u << 22) | (3u << 25));
  g1[1] = (int)((NEF & 0xFFFFu) << 16);              // tensor_dim0[15:0]=1024
  g1[2] = (int)(((NEF >> 16) & 0xFFFFu) |            // tensor_dim0[31:16]
                ((NN & 0xFFFFu) << 16));             // tensor_dim1[15:0]
  g1[3] = (int)(((NN >> 16) & 0xFFFFu) |             // tensor_dim1[31:16]
                (16u << 16));                        // tile_dim0 = 16
  g1[4] = (int)64u;                                  // tile_dim1 = 64
  g1[5] = (int)NEF;                                  // dim0_stride[31:0]=1024
  g1[6] = 0;                                         // stride hi, dim1_stride
  g1[7] = 0;
  v4i gz4 = {0, 0, 0, 0};
  v8i gz8 = {0, 0, 0, 0, 0, 0, 0, 0};
  __builtin_amdgcn_tensor_load_to_lds(g0, g1, gz4, gz4, gz8, 0);
}
#define WAIT_TENSOR(n) __builtin_amdgcn_s_wait_tensorcnt(n)
#define G1_ASYNC_OPS 4
#else
#define WAIT_TENSOR(n) do {} while (0)
#define G1_ASYNC_OPS 5
#endif

// ---------------------------------------------------------------------------
// K0a: z_proj = relu(z @ W_lat + b_lat)    [256,64]x[64,256] -> [256,256]
// ---------------------------------------------------------------------------
__global__ __launch_bounds__(256) void zproj_kernel(
    const float* __restrict__ z, const float* __restrict__ W_lat,
    const float* __restrict__ b_lat, float* __restrict__ z_proj) {
  int idx = blockIdx.x * 256 + threadIdx.x;   // 65536 outputs
  int r = idx >> 8;
  int c = idx & 255;
  float acc = b_lat[c];
  for (int k = 0; k < LL; ++k) acc += z[r * LL + k] * W_lat[k * DD2 + c];
  z_proj[idx] = fmaxf(acc, 0.0f);
}

// ---------------------------------------------------------------------------
// K0b: P_nd  = z_proj @ W_nd1                      [256,128]
//      A_edge= z_proj @ W_ed1[0:256]   + b_ed1     [256,128]
//      B_edge= z_proj @ W_ed1[256:512]             [256,128]
// ---------------------------------------------------------------------------
__global__ __launch_bounds__(256) void precomp_kernel(
    const float* __restrict__ zp, const float* __restrict__ W_nd1,
    const float* __restrict__ W_ed1, const float* __restrict__ b_ed1,
    float* __restrict__ P_nd, float* __restrict__ A_edge,
    float* __restrict__ B_edge) {
  int idx = blockIdx.x * 256 + threadIdx.x;   // 98304 outputs
  int which = idx >> 15;
  int local = idx & 32767;
  int r = local >> 7;
  int c = local & 127;
  const float* zr = zp + r * DD2;
  float acc;
  if (which == 0) {
    acc = 0.0f;
    for (int k = 0; k < DD2; ++k) acc += zr[k] * W_nd1[k * HH + c];
    P_nd[local] = acc;
  } else if (which == 1) {
    acc = b_ed1[c];
    for (int k = 0; k < DD2; ++k) acc += zr[k] * W_ed1[k * HH + c];
    A_edge[local] = acc;
  } else {
    acc = 0.0f;
    for (int k = 0; k < DD2; ++k) acc += zr[k] * W_ed1[(DD2 + k) * HH + c];
    B_edge[local] = acc;
  }
}

// ---------------------------------------------------------------------------
// Fused node kernel: per block, 64 node rows.
//   GEMM1 (WMMA): nep = relu(node_emb[64,1024] @ W_nep[1024,256] + b_nep)
//                 A tile via TDM (tensor_load_to_lds + LDS pad), B tile via
//                 async global->LDS; both double-buffered.
//   GEMM2 (WMMA): t   = nep @ W_nd1[256,128] + b_nd1 + P_nd[seg]
//   GEMM3:        recon_node = relu(t) @ W_nd2[128,4] + b_nd2
// ---------------------------------------------------------------------------
__global__ __launch_bounds__(256) void node_kernel(
    const float* __restrict__ node_emb, const int* __restrict__ seg,
    const float* __restrict__ W_nep, const float* __restrict__ b_nep,
    const float* __restrict__ W_nd1, const float* __restrict__ b_nd1,
    const float* __restrict__ W_nd2, const float* __restrict__ b_nd2,
    const float* __restrict__ P_nd, float* __restrict__ recon_node) {
  __shared__ float sNep[64 * 257];    // 65.8 KB: nep tile, later h tile
  __shared__ float sA[2][64 * 20];    // 10.2 KB: A tiles (stride 20 floats)
  __shared__ float sB[2][16 * 256];   // 32 KB  : B tiles
  __shared__ int sSeg[64];
  __shared__ float sW2[HH * 4];

  const int t = threadIdx.x;
  const int wv = t >> 5;
  const int lane = t & 31;
  const int lhalf = lane >> 4;   // 0 | 1
  const int l15 = lane & 15;
  const int m0 = blockIdx.x * 64;

  if (t < 64) sSeg[t] = seg[m0 + t];
  for (int i = t; i < HH * 4; i += 256) sW2[i] = W_nd2[i];

  const v8f zero8 = {0.f, 0.f, 0.f, 0.f, 0.f, 0.f, 0.f, 0.f};

  // ----- GEMM1: M=64, N=256, K=1024 ------------------------------------
  v8f acc1[8];
#pragma unroll
  for (int n = 0; n < 8; ++n) acc1[n] = zero8;

  const int mr1 = (wv >> 1) * 16;   // 0,16,32,48
  const int cb1 = (wv & 1) * 128;   // 0,128

  auto stageG1 = [&](int buf, int kt) {
#if HAVE_TDM
    // A tile: 64 rows x 16 floats, row stride 1024 -> one TDM descriptor
    if (wv == 0)
      tdm_load_tile_a(node_emb + (size_t)m0 * NEF + kt, lds_off(&sA[buf][0]));
#else
    const int arow = t >> 2;
    const int achk = (t & 3) * 4;
    cp16_async(node_emb + (size_t)(m0 + arow) * NEF + kt + achk,
               &sA[buf][arow * 20 + achk]);
#endif
    // B tile: 16 rows x 256 floats (4 x B128 per thread, async)
#pragma unroll
    for (int r4 = 0; r4 < 4; ++r4) {
      int row = (t >> 6) + r4 * 4;
      int col = (t & 63) * 4;
      cp16_async(W_nep + (size_t)(kt + row) * DD2 + col,
                 &sB[buf][row * DD2 + col]);
    }
  };

  stageG1(0, 0);   // prologue: fill buffer 0

  for (int kt = 0; kt < NEF; kt += 16) {
    const int cur = (kt >> 4) & 1;
    const bool has_next = (kt + 16) < NEF;
    if (has_next) {
      stageG1(cur ^ 1, kt + 16);
      WAIT_ASYNC(G1_ASYNC_OPS);   // newly issued outstanding => cur B done
#if HAVE_TDM
      if (wv == 0) WAIT_TENSOR(1);   // in-order => cur A tile done
#endif
    } else {
      WAIT_ASYNC(0);
#if HAVE_TDM
      if (wv == 0) WAIT_TENSOR(0);
#endif
    }
    __syncthreads();
    const float* bA = &sA[cur][0];
    const float* bB = &sB[cur][0];
#pragma unroll
    for (int ks = 0; ks < 4; ++ks) {
      int kb = ks * 4 + lhalf * 2;
      v2f a;
      a.x = bA[(mr1 + l15) * 20 + kb];
      a.y = bA[(mr1 + l15) * 20 + kb + 1];
#pragma unroll
      for (int n = 0; n < 8; ++n) {
        int col = cb1 + n * 16 + l15;
        v2f b;
        b.x = bB[kb * DD2 + col];
        b.y = bB[(kb + 1) * DD2 + col];
        acc1[n] = __builtin_amdgcn_wmma_f32_16x16x4_f32(
            false, a, false, b, (short)0, acc1[n], false, false);
      }
    }
    __syncthreads();
  }

  // epilogue GEMM1: bias + relu -> sNep
#pragma unroll
  for (int n = 0; n < 8; ++n) {
#pragma unroll
    for (int j = 0; j < 8; ++j) {
      int row = mr1 + j + lhalf * 8;
      int col = cb1 + n * 16 + l15;
      float v = acc1[n][j] + b_nep[col];
      sNep[row * 257 + col] = fmaxf(v, 0.0f);
    }
  }
  __syncthreads();

  // ----- GEMM2: M=64, N=128, K=256 -------------------------------------
  v8f acc2[4];
#pragma unroll
  for (int n = 0; n < 4; ++n) acc2[n] = zero8;
  const int mr2 = (wv >> 1) * 16;
  const int cb2 = (wv & 1) * 64;

  auto stageG2 = [&](int buf, int kt) {
    // W_nd1 tile: 16 rows x 128 floats (2 x B128 per thread, async)
#pragma unroll
    for (int r2 = 0; r2 < 2; ++r2) {
      int row = (t >> 5) + r2 * 8;
      int col = (t & 31) * 4;
      cp16_async(W_nd1 + (size_t)(kt + row) * HH + col,
                 &sB[buf][row * HH + col]);
    }
  };

  stageG2(0, 0);

  for (int kt = 0; kt < DD2; kt += 16) {
    const int cur = (kt >> 4) & 1;
    const bool has_next = (kt + 16) < DD2;
    if (has_next) {
      stageG2(cur ^ 1, kt + 16);
      WAIT_ASYNC(2);
    } else {
      WAIT_ASYNC(0);
    }
    __syncthreads();
    const float* bB = &sB[cur][0];
#pragma unroll
    for (int ks = 0; ks < 4; ++ks) {
      int klo = ks * 4 + lhalf * 2;
      v2f a;
      a.x = sNep[(mr2 + l15) * 257 + kt + klo];
      a.y = sNep[(mr2 + l15) * 257 + kt + klo + 1];
#pragma unroll
      for (int n = 0; n < 4; ++n) {
        int col = cb2 + n * 16 + l15;
        v2f b;
        b.x = bB[klo * HH + col];
        b.y = bB[(klo + 1) * HH + col];
        acc2[n] = __builtin_amdgcn_wmma_f32_16x16x4_f32(
            false, a, false, b, (short)0, acc2[n], false, false);
      }
    }
    __syncthreads();
  }

  // epilogue GEMM2: + b_nd1 + P_nd[seg], relu -> reuse sNep cols [0,128)
#pragma unroll
  for (int n = 0; n < 4; ++n) {
#pragma unroll
    for (int j = 0; j < 8; ++j) {
      int row = mr2 + j + lhalf * 8;
      int col = cb2 + n * 16 + l15;
      float v = acc2[n][j] + b_nd1[col] + P_nd[(size_t)sSeg[row] * HH + col];
      sNep[row * 257 + col] = fmaxf(v, 0.0f);
    }
  }
  __syncthreads();

  // ----- GEMM3: [64,128] @ [128,4] -> recon_node ------------------------
  {
    int row = t >> 2;   // 0..63
    int c = t & 3;
    float acc = b_nd2[c];
    for (int j = 0; j < HH; ++j) acc += sNep[row * 257 + j] * sW2[j * 4 + c];
    recon_node[(size_t)(m0 + row) * 4 + c] = acc;
  }
}

// ---------------------------------------------------------------------------
// Edge kernel: per edge, h = relu(A_edge[seg[src]] + B_edge[seg[dst]]),
//              recon_edge = h @ W_ed2 + b_ed2
// ---------------------------------------------------------------------------
__global__ __launch_bounds__(256) void edge_kernel(
    const int* __restrict__ src, const int* __restrict__ dst,
    const int* __restrict__ seg, const float* __restrict__ A_edge,
    const float* __restrict__ B_edge, const float* __restrict__ W_ed2,
    const float* __restrict__ b_ed2, float* __restrict__ recon_edge) {
  __shared__ float sW[HH * 3];
  __shared__ float sb[3];
  int t = threadIdx.x;
  for (int i = t; i < HH * 3; i += 256) sW[i] = W_ed2[i];
  if (t < 3) sb[t] = b_ed2[t];
  __syncthreads();

  int e = blockIdx.x * 256 + t;   // EE == gridDim*256 exactly
  int si = seg[src[e]];
  int di = seg[dst[e]];
  const float4* Ap = reinterpret_cast<const float4*>(A_edge + (size_t)si * HH);
  const float4* Bp = reinterpret_cast<const float4*>(B_edge + (size_t)di * HH);
  float a0 = sb[0], a1 = sb[1], a2 = sb[2];
#pragma unroll 4
  for (int j = 0; j < HH / 4; ++j) {
    float4 av = Ap[j];
    float4 bv = Bp[j];
    float h;
    h = fmaxf(av.x + bv.x, 0.f);
    a0 += h * sW[(4 * j + 0) * 3 + 0];
    a1 += h * sW[(4 * j + 0) * 3 + 1];
    a2 += h * sW[(4 * j + 0) * 3 + 2];
    h = fmaxf(av.y + bv.y, 0.f);
    a0 += h * sW[(4 * j + 1) * 3 + 0];
    a1 += h * sW[(4 * j + 1) * 3 + 1];
    a2 += h * sW[(4 * j + 1) * 3 + 2];
    h = fmaxf(av.z + bv.z, 0.f);
    a0 += h * sW[(4 * j + 2) * 3 + 0];
    a1 += h * sW[(4 * j + 2) * 3 + 1];
    a2 += h * sW[(4 * j + 2) * 3 + 2];
    h = fmaxf(av.w + bv.w, 0.f);
    a0 += h * sW[(4 * j + 3) * 3 + 0];
    a1 += h * sW[(4 * j + 3) * 3 + 1];
    a2 += h * sW[(4 * j + 3) * 3 + 2];
  }
  recon_edge[(size_t)e * 3 + 0] = a0;
  recon_edge[(size_t)e * 3 + 1] = a1;
  recon_edge[(size_t)e * 3 + 2] = a2;
}

// ---------------------------------------------------------------------------
// Heads: pred_energy / pred_stress  (B=256 rows, one block per row)
// ---------------------------------------------------------------------------
__global__ __launch_bounds__(128) void head_kernel(
    const float* __restrict__ z, const float* __restrict__ lat,
    const float* __restrict__ W_en1, const float* __restrict__ b_en1,
    const float* __restrict__ W_en2, const float* __restrict__ b_en2,
    const float* __restrict__ W_st1, const float* __restrict__ b_st1,
    const float* __restrict__ W_st2, const float* __restrict__ b_st2,
    float* __restrict__ pred_energy, float* __restrict__ pred_stress) {
  __shared__ float hE[HH];
  __shared__ float hS[HH];
  int r = blockIdx.x;
  int j = threadIdx.x;
  float aE = b_en1[j], aS = b_st1[j];
  for (int k = 0; k < LL; ++k) {
    float zv = z[r * LL + k];
    aE += zv * W_en1[k * HH + j];
    aS += zv * W_st1[k * HH + j];
  }
  for (int k = 0; k < 6; ++k) {
    float lv = lat[k];
    aE += lv * W_en1[(LL + k) * HH + j];
    aS += lv * W_st1[(LL + k) * HH + j];
  }
  hE[j] = fmaxf(aE, 0.f);
  hS[j] = fmaxf(aS, 0.f);
  __syncthreads();
  if (j < 2) {
    float acc = b_en2[j];
    for (int k = 0; k < HH; ++k) acc += hE[k] * W_en2[k * 2 + j];
    pred_energy[r * 2 + j] = acc;
  }
  if (j < 9) {
    float acc = b_st2[j];
    for (int k = 0; k < HH; ++k) acc += hS[k] * W_st2[k * 9 + j];
    pred_stress[r * 9 + j] = acc;
  }
}

// ---------------------------------------------------------------------------
extern "C" void kernel_launch(void* const* d_in, const int* in_sizes, int n_in,
                              void* d_out, int out_size, void* d_ws,
                              size_t ws_size, hipStream_t stream) {
  const float* z = (const float*)d_in[0];
  const float* node_emb = (const float*)d_in[1];
  const float* lattice = (const float*)d_in[2];
  const int* segment_ids = (const int*)d_in[3];
  const int* src = (const int*)d_in[4];
  const int* dst = (const int*)d_in[5];
  const float* W_lat = (const float*)d_in[6];
  const float* b_lat = (const float*)d_in[7];
  const float* W_nep = (const float*)d_in[8];
  const float* b_nep = (const float*)d_in[9];
  const float* W_nd1 = (const float*)d_in[10];
  const float* b_nd1 = (const float*)d_in[11];
  const float* W_nd2 = (const float*)d_in[12];
  const float* b_nd2 = (const float*)d_in[13];
  const float* W_ed1 = (const float*)d_in[14];
  const float* b_ed1 = (const float*)d_in[15];
  const float* W_ed2 = (const float*)d_in[16];
  const float* b_ed2 = (const float*)d_in[17];
  const float* W_en1 = (const float*)d_in[18];
  const float* b_en1 = (const float*)d_in[19];
  const float* W_en2 = (const float*)d_in[20];
  const float* b_en2 = (const float*)d_in[21];
  const float* W_st1 = (const float*)d_in[22];
  const float* b_st1 = (const float*)d_in[23];
  const float* W_st2 = (const float*)d_in[24];
  const float* b_st2 = (const float*)d_in[25];
  (void)in_sizes; (void)n_in; (void)out_size; (void)ws_size;

  float* ws = (float*)d_ws;
  float* z_proj = ws;                // 65536
  float* P_nd = ws + 65536;          // 32768
  float* A_edge = ws + 98304;        // 32768
  float* B_edge = ws + 131072;       // 32768

  float* out = (float*)d_out;
  float* recon_node = out;                      // 65536*4
  float* recon_edge = out + 262144;             // 524288*3
  float* pred_energy = out + 1835008;           // 256*2
  float* pred_stress = out + 1835520;           // 256*9

  zproj_kernel<<<256, 256, 0, stream>>>(z, W_lat, b_lat, z_proj);
  precomp_kernel<<<384, 256, 0, stream>>>(z_proj, W_nd1, W_ed1, b_ed1, P_nd,
                                          A_edge, B_edge);
  node_kernel<<<NN / 64, 256, 0, stream>>>(node_emb, segment_ids, W_nep, b_nep,
                                           W_nd1, b_nd1, W_nd2, b_nd2, P_nd,
                                           recon_node);
  edge_kernel<<<EE / 256, 256, 0, stream>>>(src, dst, segment_ids, A_edge,
                                            B_edge, W_ed2, b_ed2, recon_edge);
  head_kernel<<<BB, 128, 0, stream>>>(z, lattice, W_en1, b_en1, W_en2, b_en2,
                                      W_st1, b_st1, W_st2, b_st2, pred_energy,
                                      pred_stress);
}